// GNN_Model_11639361372710
// MI455X (gfx1250) — compile-verified
//
#include <hip/hip_runtime.h>
#include <math.h>

// GCN (3-layer, PyG GCNConv semantics) for MI455X / gfx1250.
// Edge phase is the bandwidth bottleneck (~1 GB traffic/launch, L2-resident).
// Dense X@W uses V_WMMA_F32_16X16X4_F32 (fp32 WMMA, 3 chained ops per 16-row
// tile) with weights staged zero-padded in LDS and features kept in ld-16
// zero-padded buffers so all WMMA operand loads are unconditional.

typedef __attribute__((ext_vector_type(2))) float v2f;
typedef __attribute__((ext_vector_type(8))) float v8f;

#define KIN 11

// ---------------- graph prep ----------------

__global__ void k_deg_init(float* deg, int n) {
  int i = blockIdx.x * blockDim.x + threadIdx.x;
  if (i < n) deg[i] = 1.0f;  // self-loop contributes 1 to every node's degree
}

__global__ void k_deg_acc(const long long* __restrict__ dst64, float* deg, int nE) {
  int e = blockIdx.x * blockDim.x + threadIdx.x;
  if (e < nE) atomicAdd(&deg[(int)dst64[e]], 1.0f);
}

__global__ void k_dinv(float* deg, int n) {
  int i = blockIdx.x * blockDim.x + threadIdx.x;
  if (i < n) deg[i] = rsqrtf(deg[i]);  // deg >= 1 always (self-loops)
}

// Pack int64 edge index -> int32 + precompute norm_e = dinv[src]*dinv[dst].
// Reads the 51 MB int64 index once instead of once per layer.
__global__ void k_pack(const long long* __restrict__ s64,
                       const long long* __restrict__ d64,
                       const float* __restrict__ dinv,
                       int* __restrict__ s32, int* __restrict__ d32,
                       float* __restrict__ norm, int nE) {
  int e = blockIdx.x * blockDim.x + threadIdx.x;
  if (e < nE) {
    int s = (int)s64[e], d = (int)d64[e];
    s32[e] = s; d32[e] = d;
    norm[e] = dinv[s] * dinv[d];
  }
}

// Lift X (ld KIN) into a ld-16 buffer with deterministic zero pad columns.
__global__ void k_pad_x(const float* __restrict__ X, float* __restrict__ P, int n) {
  int t = blockIdx.x * blockDim.x + threadIdx.x;
  int i = t >> 4, c = t & 15;
  if (i < n) P[(size_t)i * 16 + c] = (c < KIN) ? X[(size_t)i * KIN + c] : 0.0f;
}

// ---------------- dense GEMM via WMMA ----------------
// H[16-row tile] = A[tile, ld16 zero-padded] @ W[11 x Kout], one wave per tile.
// K padded to 12 via 3x V_WMMA_F32_16X16X4_F32.
// A 16x4 f32 layout (ISA 7.12.2): lane(0-15)=M, VGPR0={K, K+2}, VGPR1={K+1, K+3}
// split across half-waves; B (4x16) assumed transposed-mirror of A striping.
// Weights staged into LDS zero-padded -> all operand loads unconditional.
__global__ void k_gemm_wmma(const float* __restrict__ A,
                            const float* __restrict__ W, int Kout,
                            float* __restrict__ H, int nTiles) {
  __shared__ float Wp[16 * 16];
  {
    int r = threadIdx.x >> 4, cc = threadIdx.x & 15;   // 256 threads = 16x16
    Wp[threadIdx.x] = (r < KIN && cc < Kout) ? W[r * Kout + cc] : 0.0f;
  }
  __syncthreads();

  int wave = (blockIdx.x * blockDim.x + threadIdx.x) >> 5;
  if (wave >= nTiles) return;                 // wave-uniform: EXEC stays all-1s
  int lane = threadIdx.x & 31;
  int g  = lane >> 4;                         // half-wave select
  int mr = lane & 15;                         // A row / B,C column
  const float* arow = A + (size_t)(wave * 16 + mr) * 16;

  v8f c = {0.f, 0.f, 0.f, 0.f, 0.f, 0.f, 0.f, 0.f};
#pragma unroll
  for (int ch = 0; ch < 3; ++ch) {
    int k0 = ch * 4 + 2 * g;                  // this half-wave's K pair
    v2f a, b;
    a[0] = arow[k0];                          // pads are guaranteed zero
    a[1] = arow[k0 + 1];
    b[0] = Wp[k0 * 16 + mr];
    b[1] = Wp[(k0 + 1) * 16 + mr];
    // 8 args: (neg_a, A, neg_b, B, c_mod, C, reuse_a, reuse_b)
    c = __builtin_amdgcn_wmma_f32_16x16x4_f32(false, a, false, b,
                                              (short)0, c, false, false);
  }
  // D layout: VGPR v, lanes 0-15 -> M=v, lanes 16-31 -> M=v+8, N=mr
  float* hrow = H + (size_t)(wave * 16 + 8 * g) * 16 + mr;
#pragma unroll
  for (int v = 0; v < 8; ++v) hrow[(size_t)v * 16] = c[v];
}

// ---------------- aggregation ----------------

// Fold self-loop: OUT[i,c] = H[i,c] * dinv[i]^2; also rewrite pad columns to
// zero for ld-16 targets (keeps ping-pong buffers deterministically padded).
__global__ void k_agg_init(const float* __restrict__ H, const float* __restrict__ dinv,
                           float* __restrict__ OUT, int n, int Kout, int ldOut) {
  int t = blockIdx.x * blockDim.x + threadIdx.x;
  int i = t >> 4, c = t & 15;
  if (i >= n) return;
  float di = dinv[i];
  float v = (c < Kout) ? H[(size_t)i * 16 + c] * di * di : 0.0f;
  if (c < ldOut) OUT[(size_t)i * ldOut + c] = v;
}

// 16 threads per edge: uniform src/dst/norm per half-wave, contiguous 44B gathers.
__global__ void k_agg_edges(const int* __restrict__ s32, const int* __restrict__ d32,
                            const float* __restrict__ norm, const float* __restrict__ H,
                            float* OUT, int nE, int Kout, int ldOut) {
  unsigned t = blockIdx.x * blockDim.x + threadIdx.x;
  unsigned e = t >> 4; int c = t & 15;
  if (e < (unsigned)nE && c < Kout) {
    int s = s32[e], d = d32[e];
    atomicAdd(&OUT[(size_t)d * ldOut + c], H[(size_t)s * 16 + c] * norm[e]);
  }
}

// ---------------- pointwise ----------------

__global__ void k_bias_gelu(float* OUT, const float* __restrict__ b, int n, int Kout) {
  int t = blockIdx.x * blockDim.x + threadIdx.x;
  int i = t >> 4, c = t & 15;
  if (i < n && c < Kout) {                    // pads stay zero
    float x = OUT[(size_t)i * 16 + c] + b[c];
    // jax.nn.gelu default (tanh approximation)
    float u = 0.7978845608028654f * (x + 0.044715f * x * x * x);
    OUT[(size_t)i * 16 + c] = 0.5f * x * (1.0f + tanhf(u));
  }
}

__global__ void k_bias_logsoftmax(float* OUT, const float* __restrict__ b, int n, int C) {
  int i = blockIdx.x * blockDim.x + threadIdx.x;
  if (i >= n) return;
  float x[16];
  float m = -INFINITY;
  for (int c = 0; c < C; ++c) {
    x[c] = OUT[(size_t)i * C + c] + b[c];
    m = fmaxf(m, x[c]);
  }
  float s = 0.0f;
  for (int c = 0; c < C; ++c) s += expf(x[c] - m);
  float ls = logf(s);
  for (int c = 0; c < C; ++c) OUT[(size_t)i * C + c] = x[c] - m - ls;
}

// ---------------- launch ----------------

extern "C" void kernel_launch(void* const* d_in, const int* in_sizes, int n_in,
                              void* d_out, int out_size, void* d_ws, size_t ws_size,
                              hipStream_t stream) {
  const float*     X     = (const float*)d_in[0];
  const long long* ei    = (const long long*)d_in[1];   // int64 [2, E]
  const float*     W1    = (const float*)d_in[2];
  const float*     b1    = (const float*)d_in[3];
  const float*     W2    = (const float*)d_in[4];
  const float*     b2    = (const float*)d_in[5];
  const float*     W3    = (const float*)d_in[6];
  const float*     b3    = (const float*)d_in[7];
  float*           out   = (float*)d_out;

  const int n  = in_sizes[0] / KIN;   // 100000
  const int nE = in_sizes[1] / 2;     // 3200000
  const int C  = in_sizes[6] / KIN;   // 10

  const long long* src64 = ei;
  const long long* dst64 = ei + nE;

  // workspace carve-out (256B aligned): ~52 MB total
  char* ws = (char*)d_ws;
  size_t off = 0;
  auto carve = [&](size_t bytes) -> void* {
    void* p = ws + off;
    off = (off + bytes + 255) & ~(size_t)255;
    return p;
  };
  float* dinv = (float*)carve((size_t)n * 4);           // deg -> dinv in place
  int*   s32  = (int*)  carve((size_t)nE * 4);
  int*   d32  = (int*)  carve((size_t)nE * 4);
  float* norm = (float*)carve((size_t)nE * 4);
  float* bufA = (float*)carve((size_t)n * 16 * 4);      // ld 16 node features
  float* bufB = (float*)carve((size_t)n * 16 * 4);

  const int B = 256;
  const int nTiles     = (n + 15) / 16;
  const int gemmBlocks = (nTiles * 32 + B - 1) / B;                    // 1 wave/tile
  const int edgeBlocks = (int)(((long long)nE * 16 + B - 1) / B);
  const int nodeBlocks = (n * 16 + B - 1) / B;

  // graph prep
  k_deg_init<<<(n + B - 1) / B, B, 0, stream>>>(dinv, n);
  k_deg_acc <<<(nE + B - 1) / B, B, 0, stream>>>(dst64, dinv, nE);
  k_dinv    <<<(n + B - 1) / B, B, 0, stream>>>(dinv, n);
  k_pack    <<<(nE + B - 1) / B, B, 0, stream>>>(src64, dst64, dinv, s32, d32, norm, nE);

  // pad X into bufA (ld 16, zero pad cols)
  k_pad_x   <<<nodeBlocks, B, 0, stream>>>(X, bufA, n);

  // layer 1: bufA @ W1 -> bufB; aggregate -> bufA; gelu(bufA + b1)
  k_gemm_wmma <<<gemmBlocks, B, 0, stream>>>(bufA, W1, KIN, bufB, nTiles);
  k_agg_init  <<<nodeBlocks, B, 0, stream>>>(bufB, dinv, bufA, n, KIN, 16);
  k_agg_edges <<<edgeBlocks, B, 0, stream>>>(s32, d32, norm, bufB, bufA, nE, KIN, 16);
  k_bias_gelu <<<nodeBlocks, B, 0, stream>>>(bufA, b1, n, KIN);

  // layer 2: bufA @ W2 -> bufB; aggregate -> bufA; gelu(bufA + b2)
  k_gemm_wmma <<<gemmBlocks, B, 0, stream>>>(bufA, W2, KIN, bufB, nTiles);
  k_agg_init  <<<nodeBlocks, B, 0, stream>>>(bufB, dinv, bufA, n, KIN, 16);
  k_agg_edges <<<edgeBlocks, B, 0, stream>>>(s32, d32, norm, bufB, bufA, nE, KIN, 16);
  k_bias_gelu <<<nodeBlocks, B, 0, stream>>>(bufA, b2, n, KIN);

  // layer 3: bufA @ W3 -> bufB; aggregate -> d_out (ld C); log_softmax
  k_gemm_wmma <<<gemmBlocks, B, 0, stream>>>(bufA, W3, C, bufB, nTiles);
  k_agg_init  <<<nodeBlocks, B, 0, stream>>>(bufB, dinv, out, n, C, C);
  k_agg_edges <<<edgeBlocks, B, 0, stream>>>(s32, d32, norm, bufB, out, nE, C, C);
  k_bias_logsoftmax<<<(n + B - 1) / B, B, 0, stream>>>(out, b3, n, C);
}